// ProposalModule_37752762532012
// MI455X (gfx1250) — compile-verified
//
#include <hip/hip_runtime.h>
#include <hip/hip_bf16.h>
#include <math.h>

typedef __attribute__((ext_vector_type(2))) float v2f;
typedef __attribute__((ext_vector_type(8))) float v8f;

#define B_      8
#define K_      16384
#define C_      256
#define NPROP   1024
#define NSAMPLE 16
#define RADIUS  0.3f
#define NUM_CLASS 18
#define NH      12
#define NS      18
#define DOUT    119         // 2 + 3 + 2*NH + 4*NS + NUM_CLASS
#define CIN     259         // C + 3
#define CINP    264         // padded to multiple of 8 (cols 259..263 zeroed)

// d_out flat offsets (floats), in reference return order
#define OBJ_OFF   0
#define CEN_OFF   (OBJ_OFF + B_*NPROP*2)
#define HS_OFF    (CEN_OFF + B_*NPROP*3)
#define HR_OFF    (HS_OFF  + B_*NPROP*NH)
#define SS_OFF    (HR_OFF  + B_*NPROP*NH)
#define SR_OFF    (SS_OFF  + B_*NPROP*NS)
#define SEM_OFF   (SR_OFF  + B_*NPROP*NS*3)
#define FEAT_OFF  (SEM_OFF + B_*NPROP*NUM_CLASS)

#define PI_OVER_NH 0.26179938779914946f  // pi/12

__device__ __forceinline__ int imin(int a, int b) { return a < b ? a : b; }

// ---------------------------------------------------------------------------
// Kernel 1: farthest point sampling. One 1024-thread block per batch.
// Each lane owns 16 points (coords + running min-dist in registers).
// ---------------------------------------------------------------------------
__device__ __forceinline__ void wave_argmax(float& v, int& i) {
#pragma unroll
    for (int off = 16; off > 0; off >>= 1) {
        float ov = __shfl_down(v, off, 32);
        int   oi = __shfl_down(i, off, 32);
        if (ov > v || (ov == v && oi < i)) { v = ov; i = oi; }
    }
}

__global__ __launch_bounds__(1024) void fps_kernel(const float* __restrict__ xyz,
                                                   float* __restrict__ ws_newxyz) {
    __shared__ float rv[32];
    __shared__ int   ri[32];
    __shared__ float sel[3];
    __shared__ int   sfar;

    const int b   = blockIdx.x;
    const int tid = threadIdx.x;
    const int lane = tid & 31;
    const int wid  = tid >> 5;
    const float* xb = xyz + (size_t)b * K_ * 3;

    float px[16], py[16], pz[16], dist[16];
#pragma unroll
    for (int j = 0; j < 16; j++) {
        const int p = (j << 10) + tid;
        const float* q = xb + p * 3;
        px[j] = q[0]; py[j] = q[1]; pz[j] = q[2];
        dist[j] = 1e10f;
    }
    if (tid == 0) { sel[0] = px[0]; sel[1] = py[0]; sel[2] = pz[0]; }
    __syncthreads();

    float* ob = ws_newxyz + (size_t)b * NPROP * 3;
    for (int t = 0; t < NPROP; t++) {
        const float sx = sel[0], sy = sel[1], sz = sel[2];
        if (tid == 0) { ob[t*3+0] = sx; ob[t*3+1] = sy; ob[t*3+2] = sz; }

        float bestv = -1.0f; int besti = 0;
#pragma unroll
        for (int j = 0; j < 16; j++) {
            const float dx = px[j] - sx, dy = py[j] - sy, dz = pz[j] - sz;
            const float d = dx*dx + dy*dy + dz*dz;
            const float nd = fminf(dist[j], d);
            dist[j] = nd;
            if (nd > bestv) { bestv = nd; besti = (j << 10) + tid; }
        }
        wave_argmax(bestv, besti);
        if (lane == 0) { rv[wid] = bestv; ri[wid] = besti; }
        __syncthreads();
        if (wid == 0) {
            float v = rv[lane]; int i2 = ri[lane];
            wave_argmax(v, i2);
            if (lane == 0) sfar = i2;
        }
        __syncthreads();
        const int far = sfar;
        if ((far & 1023) == tid) {
            const int j = far >> 10;
            sel[0] = px[j]; sel[1] = py[j]; sel[2] = pz[j];
        }
        __syncthreads();
    }
}

// ---------------------------------------------------------------------------
// Kernel 2: ball query. One wave32 per center; chunked ballot scan keeps
// ascending-index order; early exit at NSAMPLE hits; pad with first hit.
// ---------------------------------------------------------------------------
__global__ __launch_bounds__(256) void ballquery_kernel(const float* __restrict__ xyz,
                                                        const float* __restrict__ ws_newxyz,
                                                        int* __restrict__ ws_idx) {
    const int gw   = (blockIdx.x * blockDim.x + threadIdx.x) >> 5;  // center id
    const int lane = threadIdx.x & 31;
    const int b    = gw >> 10;  // gw / NPROP
    const float cx = ws_newxyz[gw*3+0];
    const float cy = ws_newxyz[gw*3+1];
    const float cz = ws_newxyz[gw*3+2];
    const float* xb = xyz + (size_t)b * K_ * 3;
    const float r2 = RADIUS * RADIUS;
    int* row = ws_idx + (size_t)gw * NSAMPLE;

    int count = 0;
    int firstReg = -1;
    for (int base = 0; base < K_ && count < NSAMPLE; base += 32) {
        const int p = base + lane;
        const float dx = xb[p*3+0] - cx;
        const float dy = xb[p*3+1] - cy;
        const float dz = xb[p*3+2] - cz;
        const bool hit = (dx*dx + dy*dy + dz*dz) < r2;
        const unsigned mask = (unsigned)__ballot(hit);
        if (mask) {
            const int prefix = __popc(mask & ((1u << lane) - 1u));
            const int pos = count + prefix;
            if (hit && pos < NSAMPLE) {
                row[pos] = p;
                if (pos == 0) firstReg = p;
            }
            count += __popc(mask);
        }
    }
    // broadcast first hit (exactly one lane has it >= 0)
#pragma unroll
    for (int off = 1; off < 32; off <<= 1)
        firstReg = max(firstReg, __shfl_xor(firstReg, off, 32));
    const int cc = min(count, NSAMPLE);
    if (lane >= cc && lane < NSAMPLE) row[lane] = firstReg;
}

// ---------------------------------------------------------------------------
// WMMA helpers: V_WMMA_F32_16X16X4_F32, D = A(16x4) x B(4x16) + C.
// A: lane L -> m=L&15, K-pair = 2*(L>>4).   B: lane L -> n=L&15, same K-pair.
// C/D: VGPR r, lane L -> row r+8*(L>>4), col L&15.
// Branch-free boundary handling: A is zero-padded in LDS, weight index is
// CLAMPED (v_min) so loads are unconditional; clamped products hit A==0.
// Two accumulators (k-step 8) break the WMMA dependency chain.
// ---------------------------------------------------------------------------
template<bool ALIGNED>
__device__ __forceinline__ v8f mlp_tile(const float* __restrict__ Alds, int lda,
                                        const float* __restrict__ Wg, int ldw,
                                        int nbase, int kPad, int kValid, int nValid) {
    const int lane = threadIdx.x & 31;
    const int half = lane >> 4;
    const int mn   = lane & 15;           // row for A-frag, col for B-frag
    const int ch   = imin(nbase + mn, nValid - 1);
    const float* arow = Alds + mn * lda + 2 * half;
    const float* wrow = Wg + (size_t)ch * ldw;
    const int kc = kValid - 1;
    v8f acc0 = {}, acc1 = {};
#pragma unroll 4
    for (int k0 = 0; k0 < kPad; k0 += 8) {
        v2f a0, a1, w0, w1;
        a0.x = arow[k0];     a0.y = arow[k0 + 1];
        a1.x = arow[k0 + 4]; a1.y = arow[k0 + 5];
        if (ALIGNED) {
            // ch*ldw + 2*half + k0 is 8-byte aligned (ldw even, k0 % 8 == 0)
            w0 = *(const v2f*)(wrow + 2 * half + k0);
            w1 = *(const v2f*)(wrow + 2 * half + k0 + 4);
        } else {
            const int ka = k0 + 2 * half;
            w0.x = wrow[imin(ka,     kc)];
            w0.y = wrow[imin(ka + 1, kc)];
            w1.x = wrow[imin(ka + 4, kc)];
            w1.y = wrow[imin(ka + 5, kc)];
        }
        acc0 = __builtin_amdgcn_wmma_f32_16x16x4_f32(false, a0, false, w0,
                                                     (short)0, acc0, false, false);
        acc1 = __builtin_amdgcn_wmma_f32_16x16x4_f32(false, a1, false, w1,
                                                     (short)0, acc1, false, false);
    }
    v8f acc;
#pragma unroll
    for (int r = 0; r < 8; r++) acc[r] = acc0[r] + acc1[r];
    return acc;
}

__device__ __forceinline__ void store_act(v8f acc, float* __restrict__ Out, int ldo,
                                          int nbase, const float* __restrict__ g,
                                          const float* __restrict__ bia,
                                          int nValid, bool do_relu) {
    const int lane = threadIdx.x & 31;
    const int half = lane >> 4;
    const int n    = lane & 15;
    const int ch   = nbase + n;
    const int chc  = imin(ch, nValid - 1);      // clamped, branch-free
    const float gg = g ? g[chc] : 1.f;
    const float bb = bia ? bia[chc] : 0.f;
#pragma unroll
    for (int r = 0; r < 8; r++) {
        const int row = r + 8 * half;
        float v = acc[r] * gg + bb;
        if (do_relu) v = fmaxf(v, 0.f);
        Out[row * ldo + ch] = v;
    }
}

// ---------------------------------------------------------------------------
// Kernel 3: gather + 3-layer shared MLP (WMMA f32) + max-pool over samples.
// One block per proposal; 8 waves each own one 16-wide N tile.
// ---------------------------------------------------------------------------
__global__ __launch_bounds__(256) void pointnet_kernel(
        const float* __restrict__ xyz, const float* __restrict__ features,
        const float* __restrict__ ws_newxyz, const int* __restrict__ ws_idx,
        const float* __restrict__ W1, const float* __restrict__ g1, const float* __restrict__ b1,
        const float* __restrict__ W2, const float* __restrict__ g2, const float* __restrict__ b2,
        const float* __restrict__ W3, const float* __restrict__ g3, const float* __restrict__ b3,
        float* __restrict__ out_feat) {
    __shared__ float xin [NSAMPLE * CINP];
    __shared__ float buf1[NSAMPLE * 128];
    __shared__ float buf2[NSAMPLE * 128];

    const int blk = blockIdx.x;            // b*NPROP + p
    const int b   = blk >> 10;
    const int tid = threadIdx.x;
    const int* irow = ws_idx + (size_t)blk * NSAMPLE;
    const float* nx = ws_newxyz + (size_t)blk * 3;

    if (tid < NSAMPLE * 3) {
        const int s = tid / 3, c = tid % 3;
        const int gi = irow[s];
        xin[s * CINP + c] = (xyz[((size_t)b * K_ + gi) * 3 + c] - nx[c]) * (1.f / RADIUS);
    }
    if (tid < NSAMPLE * (CINP - CIN)) {    // zero the pad columns 259..263
        const int s = tid / (CINP - CIN), c = CIN + tid % (CINP - CIN);
        xin[s * CINP + c] = 0.f;
    }
    for (int e = tid; e < NSAMPLE * C_; e += 256) {
        const int s = e >> 8, c = e & 255;
        const int gi = irow[s];
        xin[s * CINP + 3 + c] = features[((size_t)b * C_ + c) * K_ + gi];
    }
    __syncthreads();

    const int nbase = (tid >> 5) * 16;
    v8f acc = mlp_tile<false>(xin, CINP, W1, CIN, nbase, CINP, CIN, 128);
    store_act(acc, buf1, 128, nbase, g1, b1, 128, true);
    __syncthreads();
    acc = mlp_tile<true>(buf1, 128, W2, 128, nbase, 128, 128, 128);
    store_act(acc, buf2, 128, nbase, g2, b2, 128, true);
    __syncthreads();
    acc = mlp_tile<true>(buf2, 128, W3, 128, nbase, 128, 128, 128);
    store_act(acc, buf1, 128, nbase, g3, b3, 128, true);
    __syncthreads();

    if (tid < 128) {
        float mx = buf1[tid];
#pragma unroll
        for (int s = 1; s < NSAMPLE; s++) mx = fmaxf(mx, buf1[s * 128 + tid]);
        out_feat[(size_t)blk * 128 + tid] = mx;
    }
}

// ---------------------------------------------------------------------------
// Kernel 4: proposal head (3 WMMA layers over 16-proposal tiles) + scatter.
// ---------------------------------------------------------------------------
__global__ __launch_bounds__(256) void head_kernel(
        const float* __restrict__ feat, const float* __restrict__ ws_newxyz,
        const float* __restrict__ P1, const float* __restrict__ pg1, const float* __restrict__ pb1,
        const float* __restrict__ P2, const float* __restrict__ pg2, const float* __restrict__ pb2,
        const float* __restrict__ P3, const float* __restrict__ pb3,
        const float* __restrict__ msa, float* __restrict__ out) {
    __shared__ float fin[16 * 128];
    __shared__ float h1 [16 * 128];
    __shared__ float h2 [16 * 128];
    __shared__ float net[16 * 128];

    const int tile = blockIdx.x;           // 16 proposals per tile
    const int tid  = threadIdx.x;
    const float* fp = feat + (size_t)tile * 16 * 128;
    for (int e = tid; e < 16 * 128; e += 256) fin[e] = fp[e];
    __syncthreads();

    const int nbase = (tid >> 5) * 16;
    v8f acc = mlp_tile<true>(fin, 128, P1, 128, nbase, 128, 128, 128);
    store_act(acc, h1, 128, nbase, pg1, pb1, 128, true);
    __syncthreads();
    acc = mlp_tile<true>(h1, 128, P2, 128, nbase, 128, 128, 128);
    store_act(acc, h2, 128, nbase, pg2, pb2, 128, true);
    __syncthreads();
    acc = mlp_tile<true>(h2, 128, P3, 128, nbase, 128, 128, DOUT);
    store_act(acc, net, 128, nbase, nullptr, pb3, DOUT, false);
    __syncthreads();

    for (int e = tid; e < 16 * DOUT; e += 256) {
        const int s = e / DOUT, j = e % DOUT;
        const int pr = tile * 16 + s;
        const float v = net[s * 128 + j];
        if (j < 2) {
            out[OBJ_OFF + pr * 2 + j] = v;
        } else if (j < 5) {
            out[CEN_OFF + pr * 3 + (j - 2)] = ws_newxyz[pr * 3 + (j - 2)] + v;
        } else if (j < 5 + NH) {
            out[HS_OFF + pr * NH + (j - 5)] = v;
        } else if (j < 5 + 2 * NH) {
            out[HR_OFF + pr * NH + (j - 5 - NH)] = v * PI_OVER_NH;
        } else if (j < 5 + 2 * NH + NS) {
            out[SS_OFF + pr * NS + (j - 5 - 2 * NH)] = v;
        } else if (j < 5 + 2 * NH + 4 * NS) {
            const int r = j - (5 + 2 * NH + NS);
            out[SR_OFF + pr * NS * 3 + r] = v * msa[r];
        } else {
            out[SEM_OFF + pr * NUM_CLASS + (j - (5 + 2 * NH + 4 * NS))] = v;
        }
    }
}

// ---------------------------------------------------------------------------
extern "C" void kernel_launch(void* const* d_in, const int* in_sizes, int n_in,
                              void* d_out, int out_size, void* d_ws, size_t ws_size,
                              hipStream_t stream) {
    const float* xyz      = (const float*)d_in[0];
    const float* features = (const float*)d_in[1];
    const float* W1 = (const float*)d_in[2];
    const float* g1 = (const float*)d_in[3];
    const float* b1 = (const float*)d_in[4];
    const float* W2 = (const float*)d_in[5];
    const float* g2 = (const float*)d_in[6];
    const float* b2 = (const float*)d_in[7];
    const float* W3 = (const float*)d_in[8];
    const float* g3 = (const float*)d_in[9];
    const float* b3 = (const float*)d_in[10];
    const float* P1 = (const float*)d_in[11];
    const float* pg1 = (const float*)d_in[12];
    const float* pb1 = (const float*)d_in[13];
    const float* P2 = (const float*)d_in[14];
    const float* pg2 = (const float*)d_in[15];
    const float* pb2 = (const float*)d_in[16];
    const float* P3 = (const float*)d_in[17];
    const float* pb3 = (const float*)d_in[18];
    const float* msa = (const float*)d_in[19];
    float* out = (float*)d_out;

    float* ws_newxyz = (float*)d_ws;
    int*   ws_idx    = (int*)((char*)d_ws + (size_t)B_ * NPROP * 3 * sizeof(float));

    fps_kernel<<<B_, 1024, 0, stream>>>(xyz, ws_newxyz);
    ballquery_kernel<<<(B_ * NPROP * 32) / 256, 256, 0, stream>>>(xyz, ws_newxyz, ws_idx);
    pointnet_kernel<<<B_ * NPROP, 256, 0, stream>>>(xyz, features, ws_newxyz, ws_idx,
                                                    W1, g1, b1, W2, g2, b2, W3, g3, b3,
                                                    out + FEAT_OFF);
    head_kernel<<<(B_ * NPROP) / 16, 256, 0, stream>>>(out + FEAT_OFF, ws_newxyz,
                                                       P1, pg1, pb1, P2, pg2, pb2,
                                                       P3, pb3, msa, out);
}